// mrcca_40037685133551
// MI455X (gfx1250) — compile-verified
//
#include <hip/hip_runtime.h>
#include <hip/hip_bf16.h>

// ---------------------------------------------------------------------------
// MI455X (gfx1250) implementation of the axial-attention transformer block.
// fp32 end-to-end using V_WMMA_F32_16X16X4_F32 (wave32). The whole problem
// (~1.8 GFLOP, all intermediates < 26 MB) is resident in the 192 MB L2, so
// it is L2-bandwidth / latency bound: fp32 WMMA is the right precision and
// register-level tile reuse (32x32 per wave) is the main traffic lever.
// ---------------------------------------------------------------------------

typedef __attribute__((ext_vector_type(2))) float v2f;
typedef __attribute__((ext_vector_type(8))) float v8f;

#define DM_    100      // model dim
#define NSDK_  2048     // NS*DK
#define MROWS_ 1024     // SZ_B * L
#define NBATCH_ 32768   // SZ_B * NS
#define DI_    400

#define WMMA4(A, B, C) \
    __builtin_amdgcn_wmma_f32_16x16x4_f32(false, (A), false, (B), (short)0, (C), false, false)

// ---------------------------------------------------------------------------
// fp32 WMMA GEMM:  C[m,n] = epi( sum_k A[m,k] * Bw[n,k] )
//   A row-major (lda), Bw is the weight stored (N,K) row-major (ldb), C (ldc).
//   Epilogue: +bias[n] -> sw*x+sb (scalar affine) -> +residual[m,n] -> relu.
//
// One wave (32 threads) computes a 32x32 macro-tile: 2 A-frags x 2 B-frags
// -> 4 v_wmma per K-step, 4 v8f accumulators. This halves redundant A/B
// L2 reads vs a 16x16-per-wave scheme and amortizes loop overhead.
// M must be a multiple of 32, K a multiple of 4; N is guarded (wave-uniform
// full-tile fast path has zero guard VALU).
//
// Fragment layouts per CDNA5 ISA §7.12.2:
//   A (16x4, MxK):  lane l<16 holds row M=l : {a.x=K0, a.y=K1};
//                   lane l>=16 holds row M=l-16 : {a.x=K2, a.y=K3}
//   B (4x16, KxN):  symmetric — lane half selects K pair, lid selects column.
//   C/D (16x16):    VGPR i, lanes 0-15 -> (M=m0+i, N=n0+lid),
//                              lanes 16-31 -> (M=m0+i+8, N=n0+lid)
// ---------------------------------------------------------------------------
__global__ void wmma_gemm_f32(const float* __restrict__ A,
                              const float* __restrict__ Bw,
                              const float* __restrict__ bias,
                              const float* __restrict__ residual,
                              const float* __restrict__ sw,
                              const float* __restrict__ sb,
                              float* __restrict__ C,
                              int N, int K, int lda, int ldb, int ldc,
                              int ldres, int do_relu)
{
    const int lane = threadIdx.x & 31;
    const int half = lane >> 4;    // selects K pair {0,1} vs {2,3}
    const int lid  = lane & 15;    // row of A frag / column of B frag
    const int n0 = blockIdx.x * 32;
    const int m0 = blockIdx.y * 32;

    const float* a0 = A + (size_t)(m0 + lid) * lda + 2 * half;
    const float* a1 = a0 + (size_t)16 * lda;
    const int nb0 = n0 + lid;
    const int nb1 = nb0 + 16;

    v8f acc00 = {}, acc01 = {}, acc10 = {}, acc11 = {};

    if (n0 + 32 <= N) {
        // ---- full-tile fast path: no guards anywhere (all of QKV) ----
        const float* b0 = Bw + (size_t)nb0 * ldb + 2 * half;
        const float* b1 = Bw + (size_t)nb1 * ldb + 2 * half;
        for (int k = 0; k < K; k += 4) {
            const v2f af0 = *reinterpret_cast<const v2f*>(a0 + k);
            const v2f af1 = *reinterpret_cast<const v2f*>(a1 + k);
            const v2f bf0 = *reinterpret_cast<const v2f*>(b0 + k);
            const v2f bf1 = *reinterpret_cast<const v2f*>(b1 + k);
            acc00 = WMMA4(af0, bf0, acc00);
            acc01 = WMMA4(af0, bf1, acc01);
            acc10 = WMMA4(af1, bf0, acc10);
            acc11 = WMMA4(af1, bf1, acc11);
        }
    } else {
        // ---- edge tiles in N (last tile of fc / FFN1 / FFN2) ----
        const bool ok0 = nb0 < N;
        const bool ok1 = nb1 < N;
        const float* b0 = Bw + (size_t)(ok0 ? nb0 : 0) * ldb + 2 * half;
        const float* b1 = Bw + (size_t)(ok1 ? nb1 : 0) * ldb + 2 * half;
        for (int k = 0; k < K; k += 4) {
            const v2f af0 = *reinterpret_cast<const v2f*>(a0 + k);
            const v2f af1 = *reinterpret_cast<const v2f*>(a1 + k);
            v2f bf0 = *reinterpret_cast<const v2f*>(b0 + k);
            v2f bf1 = *reinterpret_cast<const v2f*>(b1 + k);
            if (!ok0) { bf0.x = 0.f; bf0.y = 0.f; }
            if (!ok1) { bf1.x = 0.f; bf1.y = 0.f; }
            acc00 = WMMA4(af0, bf0, acc00);
            acc01 = WMMA4(af0, bf1, acc01);
            acc10 = WMMA4(af1, bf0, acc10);
            acc11 = WMMA4(af1, bf1, acc11);
        }
    }

    const float scale = sw ? sw[0] : 1.0f;
    const float shift = sb ? sb[0] : 0.0f;

    // Epilogue for one 16x16 D tile with row offset mt and column nc.
    auto epi = [&](const v8f& acc, int mt, int nc) {
        if (nc < N) {
            const float bn = bias ? bias[nc] : 0.0f;
#pragma unroll
            for (int i = 0; i < 8; ++i) {
                const int m = m0 + mt + i + 8 * half;
                float v = acc[i] + bn;
                v = scale * v + shift;
                if (residual) v += residual[(size_t)m * ldres + nc];
                if (do_relu)  v = v > 0.0f ? v : 0.0f;
                C[(size_t)m * ldc + nc] = v;
            }
        }
    };
    epi(acc00,  0, nb0);
    epi(acc01,  0, nb1);
    epi(acc10, 16, nb0);
    epi(acc11, 16, nb1);
}

// ---------------------------------------------------------------------------
// Axial attention over 32768 independent (H=32, W=2) tiles.
// Batch bb = b_outer*64 + ns. The (H,W) view maps flat f=h*2+w to
// (l=f>>5, dk=f&31) of the projection buffer row m=b_outer*2+l, col ns*32+dk.
// One 64-thread block per batch; Q/K/V tiles staged in LDS.
// Output is written over the Q buffer (identical addresses, per-block
// disjoint, all global reads complete before the write).
// ---------------------------------------------------------------------------
__global__ void attn_kernel(const float* __restrict__ Q,
                            const float* __restrict__ Kb,
                            const float* __restrict__ V,
                            float* __restrict__ O)
{
    __shared__ float sQ[64], sK[64], sV[64];
    const int bb      = blockIdx.x;
    const int b_outer = bb >> 6;
    const int ns      = bb & 63;
    const int t       = threadIdx.x;   // 0..63 == flat f = h*2+w

    const size_t addr = (size_t)(b_outer * 2 + (t >> 5)) * NSDK_
                      + (size_t)ns * 32 + (t & 31);
    sQ[t] = Q[addr];
    sK[t] = Kb[addr];
    sV[t] = V[addr];
    __syncthreads();

    const int h = t >> 1;
    const int w = t & 1;
    const float q = sQ[h * 2 + w];

    // max over the 33 valid logits (g!=h over H, plus 2 over W)
    float mx = -INFINITY;
    for (int g = 0; g < 32; ++g) {
        if (g == h) continue;
        const float e = q * sK[g * 2 + w];
        mx = e > mx ? e : mx;
    }
    for (int v = 0; v < 2; ++v) {
        const float e = q * sK[h * 2 + v];
        mx = e > mx ? e : mx;
    }

    float denom = 0.0f, num = 0.0f;
    for (int g = 0; g < 32; ++g) {
        if (g == h) continue;
        const float e = expf(q * sK[g * 2 + w] - mx);
        denom += e;
        num   += e * sV[g * 2 + w];
    }
    for (int v = 0; v < 2; ++v) {
        const float e = expf(q * sK[h * 2 + v] - mx);
        denom += e;
        num   += e * sV[h * 2 + v];
    }

    O[addr] = num / denom;
}

// ---------------------------------------------------------------------------
// Row LayerNorm over DM_=100 features. One 128-thread block per row.
// ---------------------------------------------------------------------------
__global__ void layernorm_kernel(const float* __restrict__ x,
                                 const float* __restrict__ g,
                                 const float* __restrict__ b,
                                 float* __restrict__ y)
{
    __shared__ float s[128];
    const int row = blockIdx.x;
    const int t   = threadIdx.x;

    const float v = (t < DM_) ? x[(size_t)row * DM_ + t] : 0.0f;
    s[t] = v;
    __syncthreads();
    for (int off = 64; off > 0; off >>= 1) {
        if (t < off) s[t] += s[t + off];
        __syncthreads();
    }
    const float mu = s[0] * (1.0f / DM_);
    __syncthreads();

    const float d = (t < DM_) ? (v - mu) : 0.0f;
    s[t] = d * d;
    __syncthreads();
    for (int off = 64; off > 0; off >>= 1) {
        if (t < off) s[t] += s[t + off];
        __syncthreads();
    }
    const float var = s[0] * (1.0f / DM_);
    const float inv = rsqrtf(var + 1e-5f);

    if (t < DM_)
        y[(size_t)row * DM_ + t] = d * inv * g[t] + b[t];
}

// ---------------------------------------------------------------------------
extern "C" void kernel_launch(void* const* d_in, const int* in_sizes, int n_in,
                              void* d_out, int out_size, void* d_ws, size_t ws_size,
                              hipStream_t stream)
{
    const float* edges  = (const float*)d_in[0];
    const float* w_qs_w = (const float*)d_in[1];
    const float* w_qs_b = (const float*)d_in[2];
    const float* w_ks_w = (const float*)d_in[3];
    const float* w_ks_b = (const float*)d_in[4];
    const float* w_vs_w = (const float*)d_in[5];
    const float* w_vs_b = (const float*)d_in[6];
    const float* qc_w   = (const float*)d_in[7];
    const float* qc_b   = (const float*)d_in[8];
    const float* kc_w   = (const float*)d_in[9];
    const float* kc_b   = (const float*)d_in[10];
    const float* vc_w   = (const float*)d_in[11];
    const float* vc_b   = (const float*)d_in[12];
    const float* fc_w   = (const float*)d_in[13];
    const float* fc_b   = (const float*)d_in[14];
    const float* ln1_g  = (const float*)d_in[15];
    const float* ln1_b  = (const float*)d_in[16];
    const float* w1_w   = (const float*)d_in[17];
    const float* w1_b   = (const float*)d_in[18];
    const float* w2_w   = (const float*)d_in[19];
    const float* w2_b   = (const float*)d_in[20];
    const float* ln2_g  = (const float*)d_in[21];
    const float* ln2_b  = (const float*)d_in[22];

    float* ws = (float*)d_ws;
    // Workspace: 3 x (1024 x 2048) fp32 = 25.2 MB.
    float* Qb = ws;
    float* Kb = ws + (size_t)MROWS_ * NSDK_;
    float* Vb = ws + 2 * (size_t)MROWS_ * NSDK_;
    // After attention, Q holds the attention output and K's buffer is dead:
    // reuse it for all small post-attention scratch (res1/ln1/h/res2).
    float* attn = Qb;
    float* res1 = Kb;                               // 1024 x 100
    float* ln1  = Kb + (size_t)MROWS_ * DM_;        // 1024 x 100
    float* hbuf = Kb + 2 * (size_t)MROWS_ * DM_;    // 1024 x 400
    float* res2 = hbuf + (size_t)MROWS_ * DI_;      // 1024 x 100

    const dim3 wblk(32);

    // --- QKV projections: M=1024, N=2048, K=100; fused bias + scalar affine
    const dim3 gqkv(NSDK_ / 32, MROWS_ / 32);
    wmma_gemm_f32<<<gqkv, wblk, 0, stream>>>(edges, w_qs_w, w_qs_b, nullptr,
        qc_w, qc_b, Qb, NSDK_, DM_, DM_, DM_, NSDK_, 0, 0);
    wmma_gemm_f32<<<gqkv, wblk, 0, stream>>>(edges, w_ks_w, w_ks_b, nullptr,
        kc_w, kc_b, Kb, NSDK_, DM_, DM_, DM_, NSDK_, 0, 0);
    wmma_gemm_f32<<<gqkv, wblk, 0, stream>>>(edges, w_vs_w, w_vs_b, nullptr,
        vc_w, vc_b, Vb, NSDK_, DM_, DM_, DM_, NSDK_, 0, 0);

    // --- attention: one 64-thread block per (b, ns) batch
    attn_kernel<<<NBATCH_, 64, 0, stream>>>(Qb, Kb, Vb, attn);

    // --- fc: M=1024, N=100, K=2048; fused bias + edges residual
    wmma_gemm_f32<<<dim3((DM_ + 31) / 32, MROWS_ / 32), wblk, 0, stream>>>(
        attn, fc_w, fc_b, edges, nullptr, nullptr, res1,
        DM_, NSDK_, NSDK_, NSDK_, DM_, DM_, 0);
    layernorm_kernel<<<MROWS_, 128, 0, stream>>>(res1, ln1_g, ln1_b, ln1);

    // --- FFN1: M=1024, N=400, K=100; fused bias + relu
    wmma_gemm_f32<<<dim3((DI_ + 31) / 32, MROWS_ / 32), wblk, 0, stream>>>(
        ln1, w1_w, w1_b, nullptr, nullptr, nullptr, hbuf,
        DI_, DM_, DM_, DM_, DI_, 0, 1);

    // --- FFN2: M=1024, N=100, K=400; fused bias + ln1 residual
    wmma_gemm_f32<<<dim3((DM_ + 31) / 32, MROWS_ / 32), wblk, 0, stream>>>(
        hbuf, w2_w, w2_b, ln1, nullptr, nullptr, res2,
        DM_, DI_, DI_, DI_, DM_, DM_, 0);
    layernorm_kernel<<<MROWS_, 128, 0, stream>>>(res2, ln2_g, ln2_b, (float*)d_out);
}